// Attention1_37383395345124
// MI455X (gfx1250) — compile-verified
//
#include <hip/hip_runtime.h>
#include <math.h>

// MI455X / gfx1250, wave32. Matmuls via V_WMMA_F32_16X16X4_F32 (fp32 WMMA),
// tile staging via GLOBAL_LOAD_ASYNC_TO_LDS_* (ASYNCcnt), weights stream via
// nontemporal stores (TH=NT) to keep k/v resident in the 192MB L2.
//
// Problem: B=4, N=2048, C=256, H=8, Dh=32, WIN=16.
// Outputs (concat in d_out): x_out [4,2048,256] f32, attn_global [4,8,2048,2048] f32.

typedef __attribute__((ext_vector_type(2))) float v2f;
typedef __attribute__((ext_vector_type(8))) float v8f;

#define BB    4
#define NN    2048
#define CC    256
#define HH    8
#define DH    32
#define SCALE 0.17677669529663687f   // 32^-0.5

// ---------------------------------------------------------------------------
// Async global->LDS copies (CDNA5 TDM-lite path, tracked by ASYNCcnt).
// Low 32 bits of a generic pointer to LDS == LDS byte offset (aperture rule).
// ---------------------------------------------------------------------------
static __device__ __forceinline__ void async_ld_b64(void* lds_dst, const void* gsrc) {
#if defined(__gfx1250__)
  unsigned loff = (unsigned)(unsigned long long)lds_dst;
  asm volatile("global_load_async_to_lds_b64 %0, %1, off"
               :: "v"(loff), "v"((unsigned long long)gsrc) : "memory");
#else
  ((float*)lds_dst)[0] = ((const float*)gsrc)[0];
  ((float*)lds_dst)[1] = ((const float*)gsrc)[1];
#endif
}

static __device__ __forceinline__ void async_ld_b128(void* lds_dst, const void* gsrc) {
#if defined(__gfx1250__)
  unsigned loff = (unsigned)(unsigned long long)lds_dst;
  asm volatile("global_load_async_to_lds_b128 %0, %1, off"
               :: "v"(loff), "v"((unsigned long long)gsrc) : "memory");
#else
  for (int i = 0; i < 4; ++i) ((float*)lds_dst)[i] = ((const float*)gsrc)[i];
#endif
}

static __device__ __forceinline__ void wait_async0() {
#if defined(__gfx1250__)
  asm volatile("s_wait_asynccnt 0x0" ::: "memory");
#endif
}

// ---------------------------------------------------------------------------
// One 16x16 f32 tile: C += A(16xK, row-major, stride lda) * B(Kx16, strides bsr/bsc)
// A fragment (16x4 f32): lanes 0-15 hold M=0..15 K={kk,kk+1}; lanes 16-31 K={kk+2,kk+3}.
// B fragment (4x16 f32): mirrored (N striped over lane&15, K split over lane halves).
// C/D (16x16 f32, v8f): VGPR r -> M=r (lanes 0-15) / M=8+r (lanes 16-31), N=lane&15.
// ---------------------------------------------------------------------------
static __device__ __forceinline__ v8f wmma_tile_f32(const float* __restrict__ Ab, int lda,
                                                    const float* __restrict__ Bb,
                                                    long bsr, long bsc, int K, v8f c) {
  const int lane = threadIdx.x & 31;
  const int half = lane >> 4;
  const int l15  = lane & 15;
  const int koff = half * 2;
  #pragma unroll 4
  for (int kk = 0; kk < K; kk += 4) {
    v2f a, b;
    a.x = Ab[l15 * lda + kk + koff];
    a.y = Ab[l15 * lda + kk + koff + 1];
    b.x = Bb[(long)(kk + koff)     * bsr + (long)l15 * bsc];
    b.y = Bb[(long)(kk + koff + 1) * bsr + (long)l15 * bsc];
    c = __builtin_amdgcn_wmma_f32_16x16x4_f32(false, a, false, b, (short)0, c, false, false);
  }
  return c;
}

// ---------------------------------------------------------------------------
// Kernel 1: qkv = x @ W_qkv  ->  q,k,v stored as [B,H,N,DH]
// x: [8192, 256]   W_qkv: [256, 768]
// Block owns one 16-row A panel (async-staged to LDS, contiguous 16KB);
// its 8 waves cover 8 adjacent N-tiles. Grid: 512 m-tiles * 6 = 3072 blocks.
// ---------------------------------------------------------------------------
__global__ void qkv_gemm_kernel(const float* __restrict__ x,
                                const float* __restrict__ Wqkv,
                                float* __restrict__ q,
                                float* __restrict__ k,
                                float* __restrict__ v) {
  __shared__ float As[16 * CC];                   // 16KB A panel
  const int tid = threadIdx.x;
  const int m0  = (blockIdx.x / 6) * 16;
  const int nb  = blockIdx.x % 6;

  const float* Ap = x + (size_t)m0 * CC;          // contiguous 4096 floats
  #pragma unroll
  for (int p = 0; p < 4; ++p)
    async_ld_b128(As + (tid + 256 * p) * 4, Ap + (tid + 256 * p) * 4);
  wait_async0();
  __syncthreads();

  const int wv = tid >> 5;
  const int n0 = (nb * 8 + wv) * 16;

  v8f c = {};
  c = wmma_tile_f32(As, CC, Wqkv + n0, 3 * CC, 1, CC, c);

  const int lane = tid & 31;
  const int half = lane >> 4;
  const int l15  = lane & 15;
  const int c3   = n0 + l15;
  const int part = c3 >> 8;                       // 0=q 1=k 2=v
  const int h    = (c3 & 255) >> 5;
  const int d    = c3 & 31;
  float* dst = (part == 0) ? q : (part == 1) ? k : v;
  #pragma unroll
  for (int r = 0; r < 8; ++r) {
    const int m = m0 + r + half * 8;              // global row in [0, 8192)
    const int b = m >> 11;
    const int n = m & (NN - 1);
    dst[(((size_t)(b * HH + h)) * NN + n) * DH + d] = c[r];
  }
}

// ---------------------------------------------------------------------------
// Kernel 2: per (b,h,row-block-of-16): global attention row strip + softmax +
// weights write + P@V, fused with the matching 16-wide local-window attention.
// blockDim = 256 (8 waves). Dynamic LDS.
// ---------------------------------------------------------------------------
__global__ void attn_kernel(const float* __restrict__ q,
                            const float* __restrict__ k,
                            const float* __restrict__ v,
                            float* __restrict__ wout,   // [B,H,N,N]
                            float* __restrict__ xc) {   // [B,N,C] combined heads
  extern __shared__ float smem[];
  float* Ssh   = smem;                 // 16*2048 = 32768 floats (logits -> P)
  float* qs    = smem + 32768;         // 512 floats (16x32 q tile)
  float* Opart = smem + 33280;         // 9*512 floats (PV partials; k-tile staging in phase 2)
  float* red   = smem + 37888;         // 256 floats (softmax reductions)
  float* Sloc  = smem + 38144;         // 256 floats (16x16 local logits)

  const int tid  = threadIdx.x;
  const int wv   = tid >> 5;
  const int lane = tid & 31;
  const int half = lane >> 4;
  const int l15  = lane & 15;

  const int bh = blockIdx.x >> 7;                 // b*H + h
  const int i0 = (blockIdx.x & 127) * 16;         // row block start
  const int b  = bh >> 3;
  const int h  = bh & 7;

  const float* qbase = q + ((size_t)bh * NN) * DH;
  const float* kbase = k + ((size_t)bh * NN) * DH;
  const float* vbase = v + ((size_t)bh * NN) * DH;

  // ---- stage q tile (16x32 = 512 contiguous floats) into LDS, async ----
  async_ld_b64(qs + tid * 2, qbase + (size_t)i0 * DH + tid * 2);
  wait_async0();
  __syncthreads();

  // ---- S = (q_blk @ K^T) * scale, 128 column tiles across 8 waves.
  //      Each wave async-stages its contiguous 2KB k-tile into its private
  //      LDS slot (Opart region, dead until phase 5), then WMMAs from LDS. ----
  float* ksh = Opart + wv * 512;
  for (int jt = wv; jt < NN / 16; jt += 8) {
    const float* ksrc = kbase + (size_t)jt * 16 * DH;   // 512 contiguous floats
    #pragma unroll
    for (int p = 0; p < 4; ++p)
      async_ld_b128(ksh + (p * 32 + lane) * 4, ksrc + (p * 32 + lane) * 4);
    wait_async0();

    v8f c = {};
    // B(kk,n) = ktile[n*32 + kk]  -> bsr=1, bsc=32 (from LDS)
    c = wmma_tile_f32(qs, DH, ksh, 1, DH, DH, c);
    #pragma unroll
    for (int r = 0; r < 8; ++r)
      Ssh[(r + half * 8) * NN + jt * 16 + l15] = c[r] * SCALE;
  }
  __syncthreads();

  // ---- softmax over each of 16 rows (16 threads per row) ----
  const int row = tid >> 4;
  const int ts  = tid & 15;
  float mx = -INFINITY;
  for (int cI = ts; cI < NN; cI += 16) mx = fmaxf(mx, Ssh[row * NN + cI]);
  red[row * 16 + ts] = mx;
  __syncthreads();
  float rmax = red[row * 16];
  #pragma unroll
  for (int i = 1; i < 16; ++i) rmax = fmaxf(rmax, red[row * 16 + i]);
  __syncthreads();
  float sum = 0.f;
  for (int cI = ts; cI < NN; cI += 16) {
    float e = __expf(Ssh[row * NN + cI] - rmax);
    Ssh[row * NN + cI] = e;
    sum += e;
  }
  red[row * 16 + ts] = sum;
  __syncthreads();
  float rsum = 0.f;
  #pragma unroll
  for (int i = 0; i < 16; ++i) rsum += red[row * 16 + i];
  const float inv = 1.f / rsum;

  // ---- normalize; stream P to weights output nontemporally (write-once) ----
  float* wrow = wout + ((size_t)bh * NN + (size_t)(i0 + row)) * NN;
  for (int cI = ts; cI < NN; cI += 16) {
    float p = Ssh[row * NN + cI] * inv;
    Ssh[row * NN + cI] = p;
    __builtin_nontemporal_store(p, wrow + cI);
  }
  __syncthreads();

  // ---- O_global partials: wave wv covers K range [wv*256, wv*256+256) ----
  {
    const float* Ab = Ssh + wv * 256;             // lda = 2048
    v8f c0 = {}, c1 = {};
    c0 = wmma_tile_f32(Ab, NN, vbase + (long)wv * 256 * DH + 0,  DH, 1, 256, c0);
    c1 = wmma_tile_f32(Ab, NN, vbase + (long)wv * 256 * DH + 16, DH, 1, 256, c1);
    #pragma unroll
    for (int r = 0; r < 8; ++r) {
      const int m = r + half * 8;
      Opart[wv * 512 + m * DH + l15]      = c0[r];
      Opart[wv * 512 + m * DH + 16 + l15] = c1[r];
    }
  }
  __syncthreads();

  // ---- local window attention (window == this 16-row block), wave 0 ----
  if (wv == 0) {
    v8f c = {};
    c = wmma_tile_f32(qs, DH, kbase + (long)i0 * DH, 1, DH, DH, c);
    #pragma unroll
    for (int r = 0; r < 8; ++r) Sloc[(r + half * 8) * 16 + l15] = c[r] * SCALE;
  }
  __syncthreads();
  if (tid < 16) {                                  // tiny 16-wide softmax
    float m2 = -INFINITY;
    #pragma unroll
    for (int j = 0; j < 16; ++j) m2 = fmaxf(m2, Sloc[tid * 16 + j]);
    float s2 = 0.f;
    #pragma unroll
    for (int j = 0; j < 16; ++j) { float e = __expf(Sloc[tid * 16 + j] - m2); Sloc[tid * 16 + j] = e; s2 += e; }
    float iv = 1.f / s2;
    #pragma unroll
    for (int j = 0; j < 16; ++j) Sloc[tid * 16 + j] *= iv;
  }
  __syncthreads();
  if (wv == 0) {                                   // O_local = P_loc @ V_blk (9th partial)
    v8f c0 = {}, c1 = {};
    c0 = wmma_tile_f32(Sloc, 16, vbase + (long)i0 * DH + 0,  DH, 1, 16, c0);
    c1 = wmma_tile_f32(Sloc, 16, vbase + (long)i0 * DH + 16, DH, 1, 16, c1);
    #pragma unroll
    for (int r = 0; r < 8; ++r) {
      const int m = r + half * 8;
      Opart[8 * 512 + m * DH + l15]      = c0[r];
      Opart[8 * 512 + m * DH + 16 + l15] = c1[r];
    }
  }
  __syncthreads();

  // ---- reduce 9 partials, store combined head output (disjoint ownership) ----
  for (int e = tid; e < 512; e += 256) {
    float acc = 0.f;
    #pragma unroll
    for (int s = 0; s < 9; ++s) acc += Opart[s * 512 + e];
    const int m = e >> 5, d = e & 31;
    xc[((size_t)b * NN + (i0 + m)) * CC + h * DH + d] = acc;
  }
}

// ---------------------------------------------------------------------------
// Kernel 3: out = xc @ W_proj + b_proj   ([8192,256] x [256,256])
// Same A-panel staging as kernel 1. Grid: 512 m-tiles * 2 = 1024 blocks.
// ---------------------------------------------------------------------------
__global__ void proj_gemm_kernel(const float* __restrict__ xc,
                                 const float* __restrict__ Wp,
                                 const float* __restrict__ bp,
                                 float* __restrict__ out) {
  __shared__ float As[16 * CC];
  const int tid = threadIdx.x;
  const int m0  = (blockIdx.x >> 1) * 16;
  const int nb  = blockIdx.x & 1;

  const float* Ap = xc + (size_t)m0 * CC;
  #pragma unroll
  for (int p = 0; p < 4; ++p)
    async_ld_b128(As + (tid + 256 * p) * 4, Ap + (tid + 256 * p) * 4);
  wait_async0();
  __syncthreads();

  const int wv = tid >> 5;
  const int n0 = (nb * 8 + wv) * 16;

  v8f c = {};
  c = wmma_tile_f32(As, CC, Wp + n0, CC, 1, CC, c);

  const int lane = tid & 31;
  const int half = lane >> 4;
  const int l15  = lane & 15;
  const float bias = bp[n0 + l15];
  #pragma unroll
  for (int r = 0; r < 8; ++r)
    out[(size_t)(m0 + r + half * 8) * CC + n0 + l15] = c[r] + bias;
}

// ---------------------------------------------------------------------------
extern "C" void kernel_launch(void* const* d_in, const int* in_sizes, int n_in,
                              void* d_out, int out_size, void* d_ws, size_t ws_size,
                              hipStream_t stream) {
  const float* x    = (const float*)d_in[0];   // [4,2048,256]
  const float* Wqkv = (const float*)d_in[1];   // [256,768]
  const float* Wp   = (const float*)d_in[2];   // [256,256]
  const float* bp   = (const float*)d_in[3];   // [256]

  float* out  = (float*)d_out;                 // x_out [4,2048,256]
  float* wout = (float*)d_out + (size_t)BB * NN * CC;  // attn weights [4,8,2048,2048]

  float* ws = (float*)d_ws;
  const size_t QKV1 = (size_t)BB * HH * NN * DH;       // 2,097,152 floats each
  float* q  = ws;
  float* k  = ws + QKV1;
  float* v  = ws + 2 * QKV1;
  float* xc = ws + 3 * QKV1;                           // [4,2048,256]

  // 16*2048 + 512 + 9*512 + 256 + 256 floats = 38400 floats
  const int SMEM_BYTES = 38400 * (int)sizeof(float);   // 153,600 B (<320KB/WGP)
  (void)hipFuncSetAttribute((const void*)attn_kernel,
                            hipFuncAttributeMaxDynamicSharedMemorySize, SMEM_BYTES);

  // K1: 512 m-tiles * 6 blocks (8 n-tiles each)
  qkv_gemm_kernel<<<3072, 256, 0, stream>>>(x, Wqkv, q, k, v);
  // K2: B*H*(N/16) = 4096 blocks
  attn_kernel<<<4096, 256, SMEM_BYTES, stream>>>(q, k, v, wout, xc);
  // K3: 512 m-tiles * 2 blocks
  proj_gemm_kernel<<<1024, 256, 0, stream>>>(xc, Wp, bp, out);
}